// PatchEncoder1D_70901320122806
// MI455X (gfx1250) — compile-verified
//
#include <hip/hip_runtime.h>
#include <hip/hip_bf16.h>

typedef __attribute__((ext_vector_type(2))) float v2f;
typedef __attribute__((ext_vector_type(8))) float v8f;

#define B_   16
#define NS_  4096
#define M_   2048
#define K_   16
#define H_   128
#define D_   64

__device__ __forceinline__ float silu_f(float x) {
    return x * (1.0f / (1.0f + __expf(-x)));
}

// ---------------------------------------------------------------------------
// Kernel 1: gate branch. One block per batch b, 128 threads (thread = h).
//   g[b][h]  = mean_s silu(s_norm[s]*Wg1[0,h] + us[b,s]*Wg1[1,h] + bg1[h])
//   gg[b][:] = g[b] @ Wg2 + bg2
//   gcontrib[b][h] = sum_j gg[b][j] * Wz1[129+j][h]      (rows 129..256 of Wz1)
// ---------------------------------------------------------------------------
__global__ __launch_bounds__(128)
void gate_kernel(const float* __restrict__ us, const float* __restrict__ s_norm,
                 const float* __restrict__ Wg1, const float* __restrict__ bg1,
                 const float* __restrict__ Wg2, const float* __restrict__ bg2,
                 const float* __restrict__ Wz1, float* __restrict__ gcontrib) {
    const int b = blockIdx.x;
    const int h = threadIdx.x;

    __shared__ float sh_s[128];
    __shared__ float sh_u[128];
    __shared__ float sh_g[128];
    __shared__ float sh_gg[128];

    const float w0 = Wg1[h];
    const float w1 = Wg1[H_ + h];
    const float b0 = bg1[h];

    float acc = 0.0f;
    for (int s0 = 0; s0 < NS_; s0 += 128) {
        sh_s[h] = s_norm[s0 + h];
        sh_u[h] = us[b * NS_ + s0 + h];
        __syncthreads();
        #pragma unroll 8
        for (int i = 0; i < 128; ++i) {
            float x = fmaf(sh_u[i], w1, fmaf(sh_s[i], w0, b0));
            acc += silu_f(x);
        }
        __syncthreads();
    }
    sh_g[h] = acc * (1.0f / (float)NS_);
    __syncthreads();

    float gg = bg2[h];
    #pragma unroll 8
    for (int j = 0; j < H_; ++j) gg = fmaf(sh_g[j], Wg2[j * H_ + h], gg);
    sh_gg[h] = gg;
    __syncthreads();

    float gc = 0.0f;
    #pragma unroll 8
    for (int j = 0; j < H_; ++j) gc = fmaf(sh_gg[j], Wz1[(129 + j) * H_ + h], gc);
    gcontrib[b * H_ + h] = gc;
}

// ---------------------------------------------------------------------------
// Kernel 2: fold W2p into head:  Wc = W2p @ Wz1[0:128,:],
//                                bias_c = b2p @ Wz1[0:128,:] + bz1
// blocks 0..127: row k of Wc.  block 128: bias_c.
// ---------------------------------------------------------------------------
__global__ __launch_bounds__(128)
void prep_kernel(const float* __restrict__ W2p, const float* __restrict__ b2p,
                 const float* __restrict__ Wz1, const float* __restrict__ bz1,
                 float* __restrict__ Wc, float* __restrict__ bias_c) {
    const int h = threadIdx.x;
    if (blockIdx.x < 128) {
        const int k = blockIdx.x;
        float acc = 0.0f;
        #pragma unroll 8
        for (int j = 0; j < H_; ++j) acc = fmaf(W2p[k * H_ + j], Wz1[j * H_ + h], acc);
        Wc[k * H_ + h] = acc;
    } else {
        float acc = bz1[h];
        #pragma unroll 8
        for (int j = 0; j < H_; ++j) acc = fmaf(b2p[j], Wz1[j * H_ + h], acc);
        bias_c[h] = acc;
    }
}

// ---------------------------------------------------------------------------
// Kernel 3: fused hot path.  One block per 16 rows of R = b*M + m (tile fully
// inside one batch since M % 16 == 0).  4 waves / block.
//   stage0: gather + 2->H silu-MLP, mean over K  -> sh_A[16][128]
//   stage1: sh_A @ Wc (WMMA f32 16x16x4) + cnorm*Wz1[128] + gcontrib + bias_c,
//           silu -> sh_Z[16][128]
//   stage2: sh_Z @ Wz2 (WMMA) + bz2 -> out
// ---------------------------------------------------------------------------
__global__ __launch_bounds__(128)
void fused_kernel(const float* __restrict__ us, const int* __restrict__ sensor_ids,
                  const float* __restrict__ rel_x, const float* __restrict__ centers,
                  const float* __restrict__ W1p, const float* __restrict__ b1p,
                  const float* __restrict__ Wz1, const float* __restrict__ Wz2,
                  const float* __restrict__ bz2, const float* __restrict__ Wc,
                  const float* __restrict__ bias_c, const float* __restrict__ gcontrib,
                  float* __restrict__ out) {
    const int tid  = threadIdx.x;
    const int wave = tid >> 5;
    const int lane = tid & 31;
    const int R0 = blockIdx.x * 16;
    const int b  = R0 / M_;
    const int m0 = R0 % M_;

    __shared__ float sh_u [16][16];
    __shared__ float sh_rx[16][16];
    __shared__ float sh_cn[16];
    __shared__ float sh_A [16][130];   // pooled_pre, padded rows (bank-conflict-free)
    __shared__ float sh_Z [16][130];   // silu(zpre)

    // ---- stage 0a: gather patch inputs ----
    for (int idx = tid; idx < 256; idx += 128) {
        const int r = idx >> 4, k = idx & 15;
        const int m = m0 + r;
        const int sid = sensor_ids[m * K_ + k];
        sh_u [r][k] = us[b * NS_ + sid];
        sh_rx[r][k] = rel_x[m * K_ + k];
    }
    if (tid < 16) {
        const float cmin = centers[0], cmax = centers[M_ - 1];
        const float c = centers[m0 + tid];
        sh_cn[tid] = 2.0f * (c - cmin) / (cmax - cmin + 1e-12f) - 1.0f;
    }
    __syncthreads();

    // ---- stage 0b: 2->H MLP + silu, mean over K.  thread = column h. ----
    {
        const int h = tid;
        const float w0 = W1p[h], w1 = W1p[H_ + h], bb = b1p[h];
        #pragma unroll 4
        for (int r = 0; r < 16; ++r) {
            float acc = 0.0f;
            #pragma unroll
            for (int k = 0; k < 16; ++k) {
                float x = fmaf(sh_u[r][k], w1, fmaf(sh_rx[r][k], w0, bb));
                acc += silu_f(x);
            }
            sh_A[r][h] = acc * (1.0f / (float)K_);
        }
    }
    __syncthreads();

    // WMMA f32 16x16x4 fragment coords (ISA 7.12.2):
    //   A: lane<16 -> (m=lane, K=kb+0/1), lane>=16 -> (m=lane-16, K=kb+2/3)
    //   B: symmetric with n = lane%16
    //   C/D: elem j -> row j + 8*(lane/16), col = lane%16
    const int mrow  = lane & 15;
    const int khalf = lane >> 4;

    // ---- stage 1: zpre = sh_A @ Wc (+extras), silu -> sh_Z.  2 col-tiles/wave ----
    #pragma unroll
    for (int cti = 0; cti < 2; ++cti) {
        const int col0 = (wave * 2 + cti) * 16;
        const int n = col0 + (lane & 15);
        v8f acc = {};
        for (int kb = 0; kb < H_; kb += 4) {
            const int kk = kb + khalf * 2;
            v2f a, bf;
            a.x  = sh_A[mrow][kk];
            a.y  = sh_A[mrow][kk + 1];
            bf.x = Wc[kk * H_ + n];
            bf.y = Wc[(kk + 1) * H_ + n];
            acc = __builtin_amdgcn_wmma_f32_16x16x4_f32(
                false, a, false, bf, (short)0, acc, false, false);
        }
        const float gcv = gcontrib[b * H_ + n];
        const float bcv = bias_c[n];
        const float wcn = Wz1[128 * H_ + n];   // c_norm row of Wz1
        #pragma unroll
        for (int j = 0; j < 8; ++j) {
            const int ml = j + khalf * 8;
            float z = acc[j] + bcv + gcv + sh_cn[ml] * wcn;
            sh_Z[ml][n] = silu_f(z);
        }
    }
    __syncthreads();

    // ---- stage 2: out = sh_Z @ Wz2 + bz2.  1 col-tile/wave (D=64=4*16) ----
    {
        const int n = wave * 16 + (lane & 15);
        v8f acc = {};
        for (int kb = 0; kb < H_; kb += 4) {
            const int kk = kb + khalf * 2;
            v2f a, bf;
            a.x  = sh_Z[mrow][kk];
            a.y  = sh_Z[mrow][kk + 1];
            bf.x = Wz2[kk * D_ + n];
            bf.y = Wz2[(kk + 1) * D_ + n];
            acc = __builtin_amdgcn_wmma_f32_16x16x4_f32(
                false, a, false, bf, (short)0, acc, false, false);
        }
        const float bzv = bz2[n];
        #pragma unroll
        for (int j = 0; j < 8; ++j) {
            const int ml = j + khalf * 8;
            out[(size_t)(R0 + ml) * D_ + n] = acc[j] + bzv;
        }
    }
}

// ---------------------------------------------------------------------------
// Launch.  Input order (setup_inputs dict):
//  0 xs(unused) 1 us 2 sensor_ids 3 rel_x 4 s_norm 5 centers 6 W1p 7 b1p
//  8 W2p 9 b2p 10 Wg1 11 bg1 12 Wg2 13 bg2 14 Wz1 15 bz1 16 Wz2 17 bz2
// ---------------------------------------------------------------------------
extern "C" void kernel_launch(void* const* d_in, const int* in_sizes, int n_in,
                              void* d_out, int out_size, void* d_ws, size_t ws_size,
                              hipStream_t stream) {
    const float* us         = (const float*)d_in[1];
    const int*   sensor_ids = (const int*)  d_in[2];
    const float* rel_x      = (const float*)d_in[3];
    const float* s_norm     = (const float*)d_in[4];
    const float* centers    = (const float*)d_in[5];
    const float* W1p = (const float*)d_in[6];
    const float* b1p = (const float*)d_in[7];
    const float* W2p = (const float*)d_in[8];
    const float* b2p = (const float*)d_in[9];
    const float* Wg1 = (const float*)d_in[10];
    const float* bg1 = (const float*)d_in[11];
    const float* Wg2 = (const float*)d_in[12];
    const float* bg2 = (const float*)d_in[13];
    const float* Wz1 = (const float*)d_in[14];
    const float* bz1 = (const float*)d_in[15];
    const float* Wz2 = (const float*)d_in[16];
    const float* bz2 = (const float*)d_in[17];
    float* out = (float*)d_out;

    // workspace layout
    float* gcontrib = (float*)d_ws;                    // B*H          = 2048 f
    float* Wc       = gcontrib + B_ * H_;              // H*H          = 16384 f
    float* bias_c   = Wc + H_ * H_;                    // H            = 128 f

    gate_kernel<<<B_, 128, 0, stream>>>(us, s_norm, Wg1, bg1, Wg2, bg2, Wz1, gcontrib);
    prep_kernel<<<129, 128, 0, stream>>>(W2p, b2p, Wz1, bz1, Wc, bias_c);
    fused_kernel<<<(B_ * M_) / 16, 128, 0, stream>>>(
        us, sensor_ids, rel_x, centers, W1p, b1p, Wz1, Wz2, bz2,
        Wc, bias_c, gcontrib, out);
    (void)in_sizes; (void)n_in; (void)out_size; (void)ws_size;
}